// Involution_46832323396109
// MI455X (gfx1250) — compile-verified
//
#include <hip/hip_runtime.h>

typedef float v2f __attribute__((ext_vector_type(2)));
typedef float v8f __attribute__((ext_vector_type(8)));

#define HW_TOT 3136   // 56*56
#define HDIM   56
#define C1     256
#define CMID   64
#define W2O    784    // 49 * 16 groups
#define KK     49
#define BN_EPSF 1e-3f
#define LDS_STRIDE 17 // pad 16->17 to reduce bank conflicts

// ---------------------------------------------------------------------------
// Kernel 1: y[b,m,p] = ACON(BN(sum_c w1[m,c] * x[b,c,p]))
// GEMM per batch: M=64 (4 waves x 16), N=16 pixels per block, K=256 (step 4).
// ---------------------------------------------------------------------------
__global__ __launch_bounds__(128) void gemm1_bn_acon_kernel(
    const float* __restrict__ x,  const float* __restrict__ w1,
    const float* __restrict__ g1, const float* __restrict__ be1,
    const float* __restrict__ mu1,const float* __restrict__ va1,
    const float* __restrict__ ap1,const float* __restrict__ ap2,
    const float* __restrict__ ab, float* __restrict__ y)
{
    const int lane = threadIdx.x & 31;
    const int wave = threadIdx.x >> 5;
    const int half = lane >> 4;     // 0: K lo pair, 1: K hi pair (A/B frags)
    const int l16  = lane & 15;
    const int b    = blockIdx.y;
    const int p0   = blockIdx.x * 16;
    const int m0   = wave * 16;     // 4 waves cover M = 0..63
    const float* xb = x + (size_t)b * C1 * HW_TOT;

    v8f acc = {};
    for (int k = 0; k < C1; k += 4) {
        const int ka = k + half * 2;
        v2f a, bb;
        // A fragment: w1 rows m0..m0+15, K slice (row-major w1[m][c])
        a.x = w1[(m0 + l16) * C1 + ka];
        a.y = w1[(m0 + l16) * C1 + ka + 1];
        // B fragment: x rows = channels (K), cols = pixels (N)
        bb.x = xb[(size_t)ka       * HW_TOT + p0 + l16];
        bb.y = xb[(size_t)(ka + 1) * HW_TOT + p0 + l16];
        acc = __builtin_amdgcn_wmma_f32_16x16x4_f32(
            false, a, false, bb, (short)0, acc, false, false);
    }

    // Epilogue: BN + ACON-C, per C/D layout row m = m0 + j + 8*half
    #pragma unroll
    for (int j = 0; j < 8; ++j) {
        const int m = m0 + j + 8 * half;
        const float inv = rsqrtf(va1[m] + BN_EPSF);
        const float sc  = g1[m] * inv;
        const float sh  = be1[m] - mu1[m] * sc;
        const float t   = acc[j] * sc + sh;
        const float d   = (ap1[m] - ap2[m]) * t;
        const float sg  = 1.0f / (1.0f + __expf(-ab[m] * d));
        y[((size_t)b * CMID + m) * HW_TOT + p0 + l16] = d * sg + ap2[m] * t;
    }
}

// ---------------------------------------------------------------------------
// Kernel 2 (fused): weight tile (784 x 16 pixels) = BN(w2 @ y) built in LDS
// via WMMA, then involution consume: out[b,c,p] = sum_k w[g*49+k,p]*x_patch.
// Weight tensor never touches HBM (saves ~78 MB of traffic).
// B fragments (y tile, 32 VGPRs) loaded ONCE and reused for all 49 M-tiles.
// ---------------------------------------------------------------------------
__global__ __launch_bounds__(128) void invo_fused_kernel(
    const float* __restrict__ x,  const float* __restrict__ y,
    const float* __restrict__ w2,
    const float* __restrict__ g2, const float* __restrict__ be2,
    const float* __restrict__ mu2,const float* __restrict__ va2,
    float* __restrict__ out)
{
    __shared__ float wlds[W2O * LDS_STRIDE];

    const int lane = threadIdx.x & 31;
    const int wave = threadIdx.x >> 5;
    const int half = lane >> 4;
    const int l16  = lane & 15;
    const int b    = blockIdx.y;
    const int p0   = blockIdx.x * 16;
    const float* yb = y + (size_t)b * CMID * HW_TOT;

    // ---- Phase 1a: preload the full B operand (64 x 16 y tile) into regs ----
    v2f bfrag[16];
    #pragma unroll
    for (int kk = 0; kk < 16; ++kk) {
        const int ka = kk * 4 + half * 2;
        bfrag[kk].x = yb[(size_t)ka       * HW_TOT + p0 + l16];
        bfrag[kk].y = yb[(size_t)(ka + 1) * HW_TOT + p0 + l16];
    }

    // ---- Phase 1b: 49 M-tiles of the weight GEMM, split across 4 waves ----
    for (int mt = wave; mt < 49; mt += 4) {
        const int m0 = mt * 16;
        v8f acc = {};
        #pragma unroll
        for (int kk = 0; kk < 16; ++kk) {
            const int ka = kk * 4 + half * 2;
            v2f a;
            a.x = w2[(m0 + l16) * CMID + ka];
            a.y = w2[(m0 + l16) * CMID + ka + 1];
            acc = __builtin_amdgcn_wmma_f32_16x16x4_f32(
                false, a, false, bfrag[kk], (short)0, acc, false, false);
        }
        #pragma unroll
        for (int j = 0; j < 8; ++j) {
            const int m = m0 + j + 8 * half;
            const float inv = rsqrtf(va2[m] + BN_EPSF);
            const float sc  = g2[m] * inv;
            const float sh  = be2[m] - mu2[m] * sc;
            wlds[m * LDS_STRIDE + l16] = acc[j] * sc + sh;
        }
    }
    __syncthreads();

    // ---- Phase 2: involution consume ----
    // 128 threads: pix = tid&15, 8 thread-columns each owning 32 channels.
    const int tid = threadIdx.x;
    const int pix = tid & 15;
    const int p   = p0 + pix;
    const int h   = p / HDIM;
    const int w   = p % HDIM;

    for (int ci = 0; ci < 32; ++ci) {
        const int c = (tid >> 4) * 32 + ci;   // 0..255
        const int g = c >> 4;
        const int base = g * KK;
        const float* xc = x + ((size_t)b * C1 + c) * HW_TOT;

        // Prefetch next channel's window (gfx1250 global_prefetch_b8 path)
        if (ci < 31) {
            const float* xn = xc + HW_TOT;  // channel c+1
            const int hpf = (h > 3) ? (h - 3) : 0;
            __builtin_prefetch(xn + hpf * HDIM + w, 0, 0);
        }

        float accu = 0.0f;
        #pragma unroll
        for (int ky = 0; ky < 7; ++ky) {
            const int hh = h + ky - 3;
            if (hh < 0 || hh >= HDIM) continue;
            #pragma unroll
            for (int kx = 0; kx < 7; ++kx) {
                const int ww = w + kx - 3;
                if (ww < 0 || ww >= HDIM) continue;
                accu = fmaf(wlds[(base + ky * 7 + kx) * LDS_STRIDE + pix],
                            xc[hh * HDIM + ww], accu);
            }
        }
        out[((size_t)b * C1 + c) * HW_TOT + p] = accu;
    }
}

// ---------------------------------------------------------------------------
extern "C" void kernel_launch(void* const* d_in, const int* in_sizes, int n_in,
                              void* d_out, int out_size, void* d_ws, size_t ws_size,
                              hipStream_t stream) {
    const float* x   = (const float*)d_in[0];
    const float* w1  = (const float*)d_in[1];
    const float* g1  = (const float*)d_in[2];
    const float* be1 = (const float*)d_in[3];
    const float* mu1 = (const float*)d_in[4];
    const float* va1 = (const float*)d_in[5];
    const float* ap1 = (const float*)d_in[6];
    const float* ap2 = (const float*)d_in[7];
    const float* ab  = (const float*)d_in[8];
    const float* w2  = (const float*)d_in[9];
    const float* g2  = (const float*)d_in[10];
    const float* be2 = (const float*)d_in[11];
    const float* mu2 = (const float*)d_in[12];
    const float* va2 = (const float*)d_in[13];
    float* out = (float*)d_out;
    float* y   = (float*)d_ws;   // 4*64*3136 floats = 3.2 MB scratch

    dim3 grid(HW_TOT / 16, 4);   // 196 pixel tiles x 4 batches
    gemm1_bn_acon_kernel<<<grid, 128, 0, stream>>>(
        x, w1, g1, be1, mu1, va1, ap1, ap2, ab, y);
    invo_fused_kernel<<<grid, 128, 0, stream>>>(
        x, y, w2, g2, be2, mu2, va2, out);
}